// SSVI_torch_83528523973073
// MI455X (gfx1250) — compile-verified
//
#include <hip/hip_runtime.h>

typedef float v2f __attribute__((ext_vector_type(2)));
typedef float v8f __attribute__((ext_vector_type(8)));

#define NDIM  3
#define NCOL  2000
#define RANK  64
#define K1    64
#define NSAMP 8192
#define LAMBD 0.01f

// One block = 2 samples. 8 waves: wave w -> sample (blk*2 + w/4), k-block 16*(w%4).
// Rank reduction done with exact-f32 V_WMMA_F32_16X16X4_F32 (A = ones, B = products):
// D[m][n] = sum_K B[K][n]  -> every lane's c[0] holds fs[kb + (lane&15)].
__global__ __launch_bounds__(256) void ssvi_main_kernel(
    const int*   __restrict__ entries,   // [NSAMP, NDIM]
    const float* __restrict__ ys,        // [NSAMP]
    const float* __restrict__ eps,       // [NDIM, NSAMP, K1, RANK]
    const float* __restrict__ means,     // [NDIM, NCOL, RANK]
    const float* __restrict__ chols,     // [NDIM, NCOL, RANK]
    float*       __restrict__ partials)  // [gridDim.x]
{
    __shared__ float m_sh[2][NDIM][RANK];
    __shared__ float s_sh[2][NDIM][RANK];   // S = L^2
    __shared__ float red[256];

    const int tid = threadIdx.x;
    const int blk = blockIdx.x;

    // ---- Phase 1: gather embeddings into LDS + KL terms -------------------
    float kl_terms = 0.0f;
    for (int i = tid; i < 2 * NDIM * RANK; i += 256) {
        int ls  = i / (NDIM * RANK);
        int rem = i - ls * (NDIM * RANK);
        int d   = rem / RANK;
        int r   = rem - d * RANK;
        int s   = blk * 2 + ls;
        int col = entries[s * NDIM + d];
        float mv = means[((size_t)d * NCOL + col) * RANK + r];
        float lv = chols[((size_t)d * NCOL + col) * RANK + r];
        float S  = lv * lv;
        m_sh[ls][d][r] = mv;
        s_sh[ls][d][r] = S;
        // kl = -0.5 * sum(1 + log(S^2) - m^2 - S^2); sign applied later
        kl_terms += 1.0f + logf(S * S) - mv * mv - S * S;
    }
    __syncthreads();

    // ---- Phase 2: stream eps, triple product, WMMA rank-reduction ---------
    const int lane = tid & 31;
    const int w    = tid >> 5;
    const int ls   = w >> 2;                 // local sample 0/1
    const int s    = blk * 2 + ls;
    const int kb   = (w & 3) << 4;           // k-block base
    const int k    = kb + (lane & 15);
    const int rg   = (lane >> 4) << 1;       // r sub-offset: 0 or 2

    const size_t slab = (size_t)K1 * RANK;   // 4096 elements per (d, sample)
    const float* e0 = eps + ((size_t)0 * NSAMP + s) * slab + (size_t)k * RANK + rg;
    const float* e1 = eps + ((size_t)1 * NSAMP + s) * slab + (size_t)k * RANK + rg;
    const float* e2 = eps + ((size_t)2 * NSAMP + s) * slab + (size_t)k * RANK + rg;

    const v2f ones = {1.0f, 1.0f};
    v8f c = {};
#pragma unroll
    for (int r0 = 0; r0 < RANK; r0 += 4) {
        const int r = r0 + rg;
        v2f ev0 = __builtin_nontemporal_load((const v2f*)(e0 + r0));
        v2f ev1 = __builtin_nontemporal_load((const v2f*)(e1 + r0));
        v2f ev2 = __builtin_nontemporal_load((const v2f*)(e2 + r0));
        v2f m0 = *(const v2f*)&m_sh[ls][0][r];
        v2f m1 = *(const v2f*)&m_sh[ls][1][r];
        v2f m2 = *(const v2f*)&m_sh[ls][2][r];
        v2f s0 = *(const v2f*)&s_sh[ls][0][r];
        v2f s1 = *(const v2f*)&s_sh[ls][1][r];
        v2f s2 = *(const v2f*)&s_sh[ls][2][r];
        v2f f0 = m0 + ev0 * s0;
        v2f f1 = m1 + ev1 * s1;
        v2f f2 = m2 + ev2 * s2;
        v2f prod = f0 * f1 * f2;
        // D = ones(16x4) x prod(4x16) + C : column n accumulates the 4
        // rank-partials held by lane pair (n, n+16).
        c = __builtin_amdgcn_wmma_f32_16x16x4_f32(
                /*neg_a=*/false, ones, /*neg_b=*/false, prod,
                /*c_mod=*/(short)0, c, /*reuse_a=*/false, /*reuse_b=*/false);
    }

    // fs for k = kb + (lane&15) is in c[0]; lanes 16..31 are duplicates.
    float lp = 0.0f;
    if (lane < 16) {
        float fs  = c[0];
        float dlt = fs - ys[s];
        lp = -0.5f * dlt * dlt;
    }

    // loss partial: -(sum lp)/K1 + (LAMBD*0.5) * sum kl_terms
    float v = -lp * (1.0f / (float)K1) + (0.5f * LAMBD) * kl_terms;

    // ---- Phase 3: deterministic block reduction ---------------------------
    red[tid] = v;
    __syncthreads();
    for (int off = 128; off > 0; off >>= 1) {
        if (tid < off) red[tid] += red[tid + off];
        __syncthreads();
    }
    if (tid == 0) partials[blk] = red[0];
}

__global__ __launch_bounds__(256) void ssvi_reduce_kernel(
    const float* __restrict__ partials, int n, float* __restrict__ out)
{
    __shared__ float red[256];
    float acc = 0.0f;
    for (int i = threadIdx.x; i < n; i += 256) acc += partials[i];
    red[threadIdx.x] = acc;
    __syncthreads();
    for (int off = 128; off > 0; off >>= 1) {
        if (threadIdx.x < off) red[threadIdx.x] += red[threadIdx.x + off];
        __syncthreads();
    }
    if (threadIdx.x == 0) out[0] = red[0];
}

extern "C" void kernel_launch(void* const* d_in, const int* in_sizes, int n_in,
                              void* d_out, int out_size, void* d_ws, size_t ws_size,
                              hipStream_t stream) {
    const int*   entries = (const int*)  d_in[0];
    const float* ys      = (const float*)d_in[1];
    const float* eps     = (const float*)d_in[2];
    const float* means   = (const float*)d_in[3];
    const float* chols   = (const float*)d_in[4];
    float* out = (float*)d_out;
    float* ws  = (float*)d_ws;          // needs NSAMP/2 floats = 16 KB

    const int nblk = NSAMP / 2;         // 4096 blocks, 2 samples each
    ssvi_main_kernel<<<nblk, 256, 0, stream>>>(entries, ys, eps, means, chols, ws);
    ssvi_reduce_kernel<<<1, 256, 0, stream>>>(ws, nblk, out);
}